// QuantumLayer_89215060672684
// MI455X (gfx1250) — compile-verified
//
#include <hip/hip_runtime.h>
#include <math.h>

typedef __attribute__((ext_vector_type(2))) float v2f;
typedef __attribute__((ext_vector_type(8))) float v8f;

#define NQ 6
#define NL 6
#define DIM 64
#define BATCH 262144

// ---------------------------------------------------------------------------
// Kernel 0 (runs once, 1 block): build W = [Re(U) | Im(U)] (64 x 128) from
// weights, exactly replaying _build_unitary_T. Gates act on the COLUMN index,
// so each thread owns one row in LDS — no synchronization needed.
// ---------------------------------------------------------------------------
__global__ __launch_bounds__(64) void build_unitary_kernel(
    const float* __restrict__ w, float* __restrict__ W)
{
    __shared__ float Ur[DIM][DIM];
    __shared__ float Ui[DIM][DIM];
    const int row = threadIdx.x;            // 0..63

    for (int c = 0; c < DIM; ++c) {
        Ur[row][c] = (c == row) ? 1.0f : 0.0f;
        Ui[row][c] = 0.0f;
    }

    #pragma unroll 1
    for (int l = 0; l < NL; ++l) {
        // single-qubit Rot(phi, theta, omega) on qubit j (column bit 5-j)
        #pragma unroll 1
        for (int j = 0; j < NQ; ++j) {
            float phi = w[(l * NQ + j) * 3 + 0];
            float th  = w[(l * NQ + j) * 3 + 1];
            float om  = w[(l * NQ + j) * 3 + 2];
            float cth = cosf(0.5f * th), sth = sinf(0.5f * th);
            float ap = -0.5f * (phi + om), am = -0.5f * (phi - om);
            float epr = cosf(ap), epi = sinf(ap);
            float emr = cosf(am), emi = sinf(am);
            // R = [[ep*c, -conj(em)*s], [em*s, conj(ep)*c]]
            float R00r =  epr * cth, R00i =  epi * cth;
            float R01r = -emr * sth, R01i =  emi * sth;
            float R10r =  emr * sth, R10i =  emi * sth;
            float R11r =  epr * cth, R11i = -epi * cth;

            int stride = 1 << (5 - j);
            #pragma unroll 1
            for (int p = 0; p < DIM / 2; ++p) {
                int lo = p & (stride - 1);
                int hb = p >> (5 - j);
                int c0 = (hb << (6 - j)) + lo;
                int c1 = c0 + stride;
                float a0r = Ur[row][c0], a0i = Ui[row][c0];
                float a1r = Ur[row][c1], a1i = Ui[row][c1];
                Ur[row][c0] = R00r * a0r - R00i * a0i + R01r * a1r - R01i * a1i;
                Ui[row][c0] = R00r * a0i + R00i * a0r + R01r * a1i + R01i * a1r;
                Ur[row][c1] = R10r * a0r - R10i * a0i + R11r * a1r - R11i * a1i;
                Ui[row][c1] = R10r * a0i + R10i * a0r + R11r * a1i + R11i * a1r;
            }
        }
        // CNOT ring: control j, target (j + r) % n
        int r = l % (NQ - 1) + 1;
        #pragma unroll 1
        for (int j = 0; j < NQ; ++j) {
            int tq = (j + r) % NQ;
            int cmask = 1 << (5 - j);
            int tmask = 1 << (5 - tq);
            #pragma unroll 1
            for (int c = 0; c < DIM; ++c) {
                if ((c & cmask) && !(c & tmask)) {
                    int c2 = c | tmask;
                    float tr = Ur[row][c]; Ur[row][c] = Ur[row][c2]; Ur[row][c2] = tr;
                    float ti = Ui[row][c]; Ui[row][c] = Ui[row][c2]; Ui[row][c2] = ti;
                }
            }
        }
    }

    #pragma unroll 1
    for (int c = 0; c < DIM; ++c) {
        W[row * 128 + c]      = Ur[row][c];
        W[row * 128 + 64 + c] = Ui[row][c];
    }
}

// ---------------------------------------------------------------------------
// Kernel 1: fused  psi-build -> psi @ [ReU|ImU] (f32 WMMA) -> |z|^2 -> Z-exp.
// 256 threads = 8 waves; each wave owns 16 batch rows.
//
// LDS (single 36.9 KB buffer, two phases):
//   phase A: B-matrix, K-pair interleaved: pair p holds {W[2p][n], W[2p+1][n]}
//            contiguously -> each B fragment is ONE 8B-aligned ds_load_b64.
//            pair-row stride 288 floats (== 32 mod 64 banks): lanes 0-15
//            (pair 2k) use banks 0-31, lanes 16-31 (pair 2k+1) banks 32-63.
//   phase B: probs tiles, row stride 66 (lane-halves hit disjoint bank ranges).
// ---------------------------------------------------------------------------
#define WPS 288                       // pair-row stride (floats)
#define PBS 66                        // probs row stride (floats)

__global__ __launch_bounds__(256) void qexpval_kernel(
    const float* __restrict__ x, const float* __restrict__ Wg,
    float* __restrict__ out)
{
    __shared__ float lds[32 * WPS];   // 36864 B; reused for probs (8*16*PBS = 8448)

    const int tid  = threadIdx.x;
    const int wave = tid >> 5;
    const int lane = tid & 31;
    const int hi   = lane >> 4;       // lane half (selects K pair)
    const int ln   = lane & 15;

    // stage W into LDS, pair-interleaved: lds[(k>>1)*WPS + 2n + (k&1)] = Wg[k][n]
    for (int i = tid; i < DIM * 128; i += 256) {
        int k = i >> 7, n = i & 127;
        lds[(k >> 1) * WPS + 2 * n + (k & 1)] = Wg[i];
    }

    const int rowbase = blockIdx.x * 128 + wave * 16;

    // Build exactly the 32 psi values this lane's A-fragments need.
    // 32-bit A 16x4 layout: row M = lane&15; VGPR v, half h -> K = 4k + 2h + v.
    v2f a[16];
    {
        const int grow = rowbase + ln;
        float cv[NQ], sv[NQ];
        #pragma unroll
        for (int q = 0; q < NQ; ++q) {
            float ang = 1.57079632679489662f * x[grow * NQ + q];
            cv[q] = cosf(ang);
            sv[q] = sinf(ang);
        }
        #pragma unroll
        for (int k = 0; k < 16; ++k) {
            #pragma unroll
            for (int v = 0; v < 2; ++v) {
                int j = 4 * k + 2 * hi + v;   // psi index; qubit q <-> bit (5-q)
                float p = ((j >> 5) & 1) ? sv[0] : cv[0];
                p *= ((j >> 4) & 1) ? sv[1] : cv[1];
                p *= ((j >> 3) & 1) ? sv[2] : cv[2];
                p *= ((j >> 2) & 1) ? sv[3] : cv[3];
                p *= ((j >> 1) & 1) ? sv[4] : cv[4];
                p *= ( j       & 1) ? sv[5] : cv[5];
                if (v == 0) a[k].x = p; else a[k].y = p;
            }
        }
    }
    __syncthreads();   // W staged

    // GEMM: 4 column-tile pairs (Re tile t, Im tile t+4), K = 64 in 16 steps.
    // Probs are kept in registers so the LDS buffer stays free for phase B.
    float pr[4][8];
    #pragma unroll
    for (int t = 0; t < 4; ++t) {
        v8f cre = {};
        v8f cim = {};
        #pragma unroll
        for (int k = 0; k < 16; ++k) {
            int pairbase = (2 * k + hi) * WPS;         // K pair {4k+2h, 4k+2h+1}
            v2f br = *(const v2f*)&lds[pairbase + 2 * (16 * t + ln)];
            v2f bi = *(const v2f*)&lds[pairbase + 2 * (64 + 16 * t + ln)];
            cre = __builtin_amdgcn_wmma_f32_16x16x4_f32(
                      false, a[k], false, br, (short)0, cre, false, false);
            cim = __builtin_amdgcn_wmma_f32_16x16x4_f32(
                      false, a[k], false, bi, (short)0, cim, false, false);
        }
        #pragma unroll
        for (int r = 0; r < 8; ++r) {
            float zr = cre[r], zi = cim[r];
            pr[t][r] = zr * zr + zi * zi;
        }
    }
    __syncthreads();   // all waves done reading W -> reuse buffer for probs

    // D layout: VGPR r -> M = r + 8*hi, N = lane&15; column = 16t + N
    #pragma unroll
    for (int t = 0; t < 4; ++t)
        #pragma unroll
        for (int r = 0; r < 8; ++r)
            lds[(wave * 16 + r + 8 * hi) * PBS + 16 * t + ln] = pr[t][r];
    __syncthreads();

    // Z expectations: out[m,q] = sum_c probs[m][c] * (1 - 2*((c>>(5-q))&1)).
    // Rotate the column sweep per lane to spread LDS banks.
    for (int tt = lane; tt < 96; tt += 32) {
        int m = tt / 6, q = tt % 6;
        int shift = 5 - q;
        float s = 0.0f;
        for (int i = 0; i < 64; ++i) {
            int c = (i + 2 * lane) & 63;
            float p = lds[(wave * 16 + m) * PBS + c];
            s += ((c >> shift) & 1) ? -p : p;
        }
        out[(rowbase + m) * NQ + q] = s;
    }
}

extern "C" void kernel_launch(void* const* d_in, const int* in_sizes, int n_in,
                              void* d_out, int out_size, void* d_ws, size_t ws_size,
                              hipStream_t stream) {
    const float* x = (const float*)d_in[0];   // (262144, 6)
    const float* w = (const float*)d_in[1];   // (6, 6, 3)
    float* out = (float*)d_out;               // (262144, 6)
    float* W   = (float*)d_ws;                // 64 x 128 staging for [ReU|ImU]

    build_unitary_kernel<<<1, 64, 0, stream>>>(w, W);
    qexpval_kernel<<<BATCH / 128, 256, 0, stream>>>(x, W, out);
}